// MinimalConvWTA_LIF_17059610100019
// MI455X (gfx1250) — compile-verified
//
#include <hip/hip_runtime.h>

typedef float v2f __attribute__((ext_vector_type(2)));
typedef float v8f __attribute__((ext_vector_type(8)));

#define B_BATCH 256
#define T_LEN   16384
#define KBR     3        // branches
#define ALPHA   0.95f
#define TS      2048     // timesteps per conv block
#define CH      256      // timesteps per scan chunk
#define NCHUNK  (T_LEN / CH)          // 64
#define SLOTS   (CH * KBR * 32)      // floats per chunk buffer (24576 = 96 KB)
#define UNITS   (96 * (CH / 32))     // (32 b * 3 j) rows x 8 groups of 32 t = 768

// Padded-to-32 kernel weights with 1/sqrt(k) folded in.
// u[t] = sum_{i=0..31} W[i] * x[t-31+i]  ==  causal conv with left zero-pad.
__device__ __forceinline__ float wpad(int i, int n,
                                      const float* __restrict__ w0,
                                      const float* __restrict__ w1,
                                      const float* __restrict__ w2) {
  if (n == 0) return (i >= 24) ? w0[i - 24] * 0.35355339059327373f : 0.0f; // k=8
  if (n == 1) return (i >= 16) ? w1[i - 16] * 0.25f : 0.0f;                 // k=16
  if (n == 2) return w2[i] * 0.17677669529663687f;                          // k=32
  return 0.0f;
}

__global__ __launch_bounds__(256) void conv_wmma_kernel(
    const float* __restrict__ x, const float* __restrict__ w0,
    const float* __restrict__ w1, const float* __restrict__ w2,
    float* __restrict__ u_out) {
  __shared__ float xl[TS + 32];
  const int seg  = blockIdx.x;          // which TS-span of T
  const int b    = blockIdx.y;          // batch row
  const int tid  = threadIdx.x;
  const int lane = tid & 31;
  const int wave = tid >> 5;
  const int hi   = lane >> 4;           // 0: lanes 0-15, 1: lanes 16-31
  const int r    = lane & 15;           // M for A / N for B,C

  // Stage x[seg*TS - 31 .. seg*TS + TS - 1] into LDS (zero left pad)
  const float* xrow = x + (size_t)b * T_LEN;
  const int base = seg * TS - 31;
  for (int i = tid; i < TS + 31; i += 256) {
    const int t = base + i;
    xl[i] = (t >= 0) ? xrow[t] : 0.0f;
  }
  __syncthreads();

  // B-matrix (4x16 f32) register images for the 8 K-slices; constant per wave.
  // Lane L: rows {0,1} for L<16, {2,3} for L>=16; column N = L%16.
  v2f barr[8];
#pragma unroll
  for (int q = 0; q < 8; ++q) {
    const int i0 = q * 4 + hi * 2;
    barr[q].x = wpad(i0,     r, w0, w1, w2);
    barr[q].y = wpad(i0 + 1, r, w0, w1, w2);
  }

  // Each wave owns TS/(16*8) = 16 tiles of 16 timesteps.
  for (int tt = 0; tt < TS / (16 * 8); ++tt) {
    const int tile  = wave * (TS / (16 * 8)) + tt;
    const int tbase = tile * 16;
    v8f c = {};
#pragma unroll
    for (int q = 0; q < 8; ++q) {
      // A (16x4 f32): A[m][k] = xl[tbase + m + q*4 + k], m = r, k per lane-half
      const int idx = tbase + r + q * 4 + hi * 2;
      v2f a;
      a.x = xl[idx];
      a.y = xl[idx + 1];
      c = __builtin_amdgcn_wmma_f32_16x16x4_f32(false, a, false, barr[q],
                                                (short)0, c, false, false);
    }
    // D element (M = v + 8*hi, N = r); columns N<3 hold the 3 branches.
    if (r < KBR) {
      const int t0 = seg * TS + tbase + hi * 8;
      float* dst = u_out + ((size_t)b * KBR + r) * T_LEN + t0;
      *(float4*)(dst)     = make_float4(c[0], c[1], c[2], c[3]);
      *(float4*)(dst + 4) = make_float4(c[4], c[5], c[6], c[7]);
    }
  }
}

// Double-buffered, pipelined LIF-WTA scan.
// Wave 0 runs the serial recurrence on chunk c (LDS buffer p = c&1, layout [t][j][b]);
// waves 1..7 simultaneously drain spikes of chunk c-1 and async-refill u of chunk
// c+1 into buffer 1-p via GLOBAL_LOAD_ASYNC_TO_LDS_B32 (per-lane LDS scatter,
// coalesced 128B global reads, tracked on ASYNCcnt).
__global__ __launch_bounds__(256) void lif_wta_scan_kernel(
    const float* __restrict__ u, float* __restrict__ s_out) {
  extern __shared__ float buf[];        // 2 * SLOTS floats = 192 KB
  const int g    = blockIdx.x;          // batch group: b = g*32 + lane
  const int tid  = threadIdx.x;
  const int lane = tid & 31;
  const int wave = tid >> 5;
  float* buf0 = buf;
  float* buf1 = buf + SLOTS;

  // ---- Preload chunk 0 into buf0 (all 8 waves, async copies) ----
  for (int unit = wave; unit < UNITS; unit += 8) {
    const int row = unit >> 3, grp = unit & 7;  // row: b*3+j, grp: 32-t group
    const int bl = row / 3, j = row - bl * 3;
    const int tl = grp * 32 + lane;
    const unsigned long long gaddr = (unsigned long long)(const void*)(
        u + (((size_t)(g * 32 + bl) * KBR + j) * T_LEN + tl));
    const unsigned loff = (unsigned)(size_t)&buf0[(tl * 3 + j) * 32 + bl];
    asm volatile("global_load_async_to_lds_b32 %0, %1, off"
                 :: "v"(loff), "v"(gaddr) : "memory");
  }
  asm volatile("s_wait_asynccnt 0x0" ::: "memory");
  __syncthreads();

  float v0 = 0.0f, v1 = 0.0f, v2 = 0.0f;   // LIF state (wave 0 only)

  for (int c = 0; c < NCHUNK; ++c) {
    float* cur = (c & 1) ? buf1 : buf0;     // chunk c (consumed by wave 0)
    float* oth = (c & 1) ? buf0 : buf1;     // spikes of c-1 / refill for c+1

    if (wave == 0) {
      // Serial scan: one batch per lane, overwrite u with s in place.
      for (int tl = 0; tl < CH; ++tl) {
        const float u0 = cur[(tl * 3 + 0) * 32 + lane];
        const float u1 = cur[(tl * 3 + 1) * 32 + lane];
        const float u2 = cur[(tl * 3 + 2) * 32 + lane];
        v0 = fmaf(ALPHA, v0, u0);
        v1 = fmaf(ALPHA, v1, u1);
        v2 = fmaf(ALPHA, v2, u2);
        const float m0 = v0 - 1.0f, m1 = v1 - 1.0f, m2 = v2 - 1.0f;
        int jm = 0; float best = m0;
        if (m1 > best) { best = m1; jm = 1; }   // strict > keeps first-max (argmax tie rule)
        if (m2 > best) { best = m2; jm = 2; }
        const bool fire = (best >= 0.0f);       // winner spikes iff its m >= 0
        const float s0 = (fire && jm == 0) ? 1.0f : 0.0f;
        const float s1 = (fire && jm == 1) ? 1.0f : 0.0f;
        const float s2 = (fire && jm == 2) ? 1.0f : 0.0f;
        v0 -= s0; v1 -= s1; v2 -= s2;           // soft reset, THETA = 1
        cur[(tl * 3 + 0) * 32 + lane] = s0;
        cur[(tl * 3 + 1) * 32 + lane] = s1;
        cur[(tl * 3 + 2) * 32 + lane] = s2;
      }
    } else {
      // Loader waves: per unit, the SAME lane reads the spike slot (chunk c-1),
      // stores it coalesced, then async-refills the identical slot with chunk c+1.
      const int wl = wave - 1;                  // 0..6
      for (int unit = wl; unit < UNITS; unit += 7) {
        const int row = unit >> 3, grp = unit & 7;
        const int bl = row / 3, j = row - bl * 3;
        const int tl = grp * 32 + lane;
        const size_t gbase = ((size_t)(g * 32 + bl) * KBR + j) * T_LEN + tl;
        const int slot = (tl * 3 + j) * 32 + bl;
        if (c >= 1) {
          s_out[gbase + (size_t)(c - 1) * CH] = oth[slot];  // dscnt drained before store
        }
        if (c + 1 < NCHUNK) {
          const unsigned long long gaddr = (unsigned long long)(const void*)(
              u + gbase + (size_t)(c + 1) * CH);
          const unsigned loff = (unsigned)(size_t)&oth[slot];
          asm volatile("global_load_async_to_lds_b32 %0, %1, off"
                       :: "v"(loff), "v"(gaddr) : "memory");
        }
      }
      if (c + 1 < NCHUNK) {
        asm volatile("s_wait_asynccnt 0x0" ::: "memory");
      }
    }
    __syncthreads();
  }

  // ---- Drain spikes of the final chunk (all 256 threads, coalesced float4) ----
  {
    const int t0 = (NCHUNK - 1) * CH;
    float* last = ((NCHUNK - 1) & 1) ? buf1 : buf0;
    for (int idx = tid; idx < 96 * (CH / 4); idx += 256) {
      const int row = idx / (CH / 4);
      const int f   = idx - row * (CH / 4);
      const int bl  = row / 3;
      const int j   = row - bl * 3;
      const int tl  = f * 4;
      float4 val;
      val.x = last[((tl + 0) * 3 + j) * 32 + bl];
      val.y = last[((tl + 1) * 3 + j) * 32 + bl];
      val.z = last[((tl + 2) * 3 + j) * 32 + bl];
      val.w = last[((tl + 3) * 3 + j) * 32 + bl];
      *(float4*)(s_out + (((size_t)(g * 32 + bl) * KBR + j) * T_LEN + t0 + f * 4)) = val;
    }
  }
}

extern "C" void kernel_launch(void* const* d_in, const int* in_sizes, int n_in,
                              void* d_out, int out_size, void* d_ws, size_t ws_size,
                              hipStream_t stream) {
  const float* x  = (const float*)d_in[0];
  const float* w0 = (const float*)d_in[1];
  const float* w1 = (const float*)d_in[2];
  const float* w2 = (const float*)d_in[3];
  // d_in[4] (labels y) unused by the forward reference outputs

  float* u_out = (float*)d_out;
  float* s_out = u_out + (size_t)B_BATCH * KBR * T_LEN;

  dim3 grid_conv(T_LEN / TS, B_BATCH);   // 8 x 256 blocks
  conv_wmma_kernel<<<grid_conv, 256, 0, stream>>>(x, w0, w1, w2, u_out);

  const int lds_bytes = 2 * SLOTS * (int)sizeof(float);   // 192 KB
  (void)hipFuncSetAttribute((const void*)lif_wta_scan_kernel,
                            hipFuncAttributeMaxDynamicSharedMemorySize, lds_bytes);
  lif_wta_scan_kernel<<<B_BATCH / 32, 256, lds_bytes, stream>>>(u_out, s_out);
}